// Direct_Mamba_v1_90211493085234
// MI455X (gfx1250) — compile-verified
//
#include <hip/hip_runtime.h>
#include <hip/hip_bf16.h>

#define D_MODEL 512
#define D_INNER 2048
#define D_STATE 16
#define D_CONV  4
#define DT_RANK 32
#define XDBL_E  64      // DT_RANK + 2*D_STATE
#define NB      2       // batch
#define LL      4096    // sequence length

// scan decomposition: 4 lanes per channel, 4 states per lane
#define LPC     4
#define SPL     4       // D_STATE / LPC
#define RPL     8       // DT_RANK / LPC
#define CPB     64      // channels per 256-thread block
#define CHUNK   16      // timesteps staged per LDS chunk

typedef __attribute__((ext_vector_type(2))) float v2f;
typedef __attribute__((ext_vector_type(8))) float v8f;
typedef __attribute__((ext_vector_type(4))) unsigned int u32x4;
typedef __attribute__((ext_vector_type(8))) int i32x8;
typedef __attribute__((ext_vector_type(4))) int i32x4;

#if __has_builtin(__builtin_amdgcn_tensor_load_to_lds) && __has_builtin(__builtin_amdgcn_s_wait_tensorcnt)
#define USE_TDM 1
#else
#define USE_TDM 0
#endif

// Exact-precision FP32 WMMA: D(16x16) = A(16x4) * B(4x16) + C
__device__ __forceinline__ v8f wmma4(v2f a, v2f b, v8f c) {
    return __builtin_amdgcn_wmma_f32_16x16x4_f32(
        false, a, false, b, (short)0, c, false, false);
}

__device__ __forceinline__ float siluf(float x) {
    return x / (1.0f + __expf(-x));
}

__device__ __forceinline__ float fast_exp2(float x) {
#if __has_builtin(__builtin_amdgcn_exp2f)
    return __builtin_amdgcn_exp2f(x);
#else
    return exp2f(x);
#endif
}

#if USE_TDM
// Issue one TDM load (Tensor Data Mover, D# per CDNA5 ISA ch.8).
// 2D tile of 4-byte elements: tile0 x tile1, row stride stride0 (elements).
// pad_en inserts 1 dword of LDS padding every (2<<pad_code) dwords.
__device__ __forceinline__ void tdm_load(unsigned lds_off, unsigned long long ga,
                                         unsigned tile0, unsigned tile1,
                                         unsigned long long stride0,
                                         bool pad_en, unsigned pad_code) {
    u32x4 g0;
    g0.x = 1u;                                            // count=1 (valid D#)
    g0.y = lds_off;                                       // LDS byte address
    g0.z = (unsigned)ga;                                  // global_addr[31:0]
    g0.w = (unsigned)((ga >> 32) & 0x01FFFFFFull) | (2u << 30);  // addr[56:32] | type=2
    i32x8 g1;
    unsigned w0 = (2u << 16)                               // data_size = 4 bytes
                | (pad_en ? ((1u << 20) | (pad_code << 22)) : 0u); // pad_amount=0 -> 1 dw
    const unsigned td0 = 0x7FFFFFFFu, td1 = 0x7FFFFFFFu;   // huge tensor dims: no OOB
    g1[0] = (int)w0;
    g1[1] = (int)((td0 & 0xFFFFu) << 16);                  // atomic_addr=0 | td0.lo
    g1[2] = (int)(((td0 >> 16) & 0xFFFFu) | ((td1 & 0xFFFFu) << 16));
    g1[3] = (int)(((td1 >> 16) & 0xFFFFu) | ((tile0 & 0xFFFFu) << 16));
    g1[4] = (int)(tile1 & 0xFFFFu);                        // tile_dim2 = 0
    g1[5] = (int)(unsigned)(stride0 & 0xFFFFFFFFull);
    g1[6] = (int)(unsigned)((stride0 >> 32) & 0xFFFFull);  // dim1_stride = 0
    g1[7] = 0;
    i32x4 z4 = {0, 0, 0, 0};
    i32x8 z8 = {0, 0, 0, 0, 0, 0, 0, 0};
    __builtin_amdgcn_tensor_load_to_lds(g0, g1, z4, z4, z8, 0);
}
#endif

// -------------------------------------------------------------------------
// Kernel 1: zx[b,e,l] = sum_d hs[b,l,d] * ipw[e,d]    (e in [0,4096))
// -------------------------------------------------------------------------
__global__ __launch_bounds__(256) void k_inproj(const float* __restrict__ hs,
                                                const float* __restrict__ ipw,
                                                float* __restrict__ zx) {
    const int lane = threadIdx.x & 31;
    const int wv   = threadIdx.x >> 5;
    const int W    = blockIdx.x * 8 + wv;
    const int LT   = LL / 32;                    // 128
    const int ET   = (2 * D_INNER) / 32;         // 128
    const int b    = W / (ET * LT);
    const int r    = W % (ET * LT);
    const int e0   = (r / LT) * 32;
    const int l0   = (r % LT) * 32;

    const int m  = lane & 15;
    const int kq = (lane >> 4) * 2;

    v8f acc00 = {}, acc01 = {}, acc10 = {}, acc11 = {};
    const float* Ar0 = ipw + (size_t)(e0 + m)      * D_MODEL;
    const float* Ar1 = ipw + (size_t)(e0 + 16 + m) * D_MODEL;
    const float* Br0 = hs  + ((size_t)b * LL + l0 + m)      * D_MODEL;
    const float* Br1 = hs  + ((size_t)b * LL + l0 + 16 + m) * D_MODEL;

    for (int k = 0; k < D_MODEL; k += 4) {
        const int kb = k + kq;
        v2f a0 = *(const v2f*)(Ar0 + kb);
        v2f a1 = *(const v2f*)(Ar1 + kb);
        v2f b0 = *(const v2f*)(Br0 + kb);
        v2f b1 = *(const v2f*)(Br1 + kb);
        acc00 = wmma4(a0, b0, acc00);
        acc01 = wmma4(a0, b1, acc01);
        acc10 = wmma4(a1, b0, acc10);
        acc11 = wmma4(a1, b1, acc11);
    }
    const int n  = lane & 15;
    const int mb = (lane >> 4) * 8;
    const size_t base = (size_t)b * (2 * D_INNER) * LL;
#pragma unroll
    for (int rr = 0; rr < 8; ++rr) {
        const int mm = mb + rr;
        zx[base + (size_t)(e0 + mm)      * LL + (l0 + n)]      = acc00[rr];
        zx[base + (size_t)(e0 + mm)      * LL + (l0 + 16 + n)] = acc01[rr];
        zx[base + (size_t)(e0 + 16 + mm) * LL + (l0 + n)]      = acc10[rr];
        zx[base + (size_t)(e0 + 16 + mm) * LL + (l0 + 16 + n)] = acc11[rr];
    }
}

// -------------------------------------------------------------------------
// Kernel 2: depthwise causal conv-4 + bias + SiLU (branch 1 in flipped time)
// -------------------------------------------------------------------------
__global__ __launch_bounds__(256) void k_conv(const float* __restrict__ zx,
                                              const float* __restrict__ cw,
                                              const float* __restrict__ cb,
                                              float* __restrict__ xc) {
    const size_t idx = (size_t)blockIdx.x * 256 + threadIdx.x;
    const int l = (int)(idx % LL);
    const int d = (int)((idx / LL) % D_INNER);
    const int b = (int)((idx / ((size_t)LL * D_INNER)) % NB);
    const int s = (int)( idx / ((size_t)LL * D_INNER * NB));

    const float* wrow = cw + (size_t)(s * D_INNER + d) * D_CONV;
    const float* xrow = zx + ((size_t)b * (2 * D_INNER) + d) * LL;
    float acc = cb[s * D_INNER + d];
#pragma unroll
    for (int k = 0; k < D_CONV; ++k) {
        const int j = l - (D_CONV - 1) + k;
        if (j >= 0) {
            const int lg = s ? (LL - 1 - j) : j;
            acc += wrow[k] * xrow[lg];
        }
    }
    xc[(((size_t)s * NB + b) * D_INNER + d) * LL + l] = siluf(acc);
}

// -------------------------------------------------------------------------
// Kernel 3: x_dbl[sb, l, e] = sum_d xpw[s, e, d] * xc[sb, d, l]  (e in [0,64))
// -------------------------------------------------------------------------
__global__ __launch_bounds__(256) void k_xdbl(const float* __restrict__ xc,
                                              const float* __restrict__ xpw,
                                              float* __restrict__ xdbl) {
    const int lane = threadIdx.x & 31;
    const int wv   = threadIdx.x >> 5;
    const int W    = blockIdx.x * 8 + wv;
    const int NT   = LL / 16;                    // 256
    const int sb   = W / NT;
    const int l0   = (W % NT) * 16;
    const int s    = sb >> 1;

    const int m  = lane & 15;
    const int kq = (lane >> 4) * 2;

    v8f acc[4] = {{}, {}, {}, {}};
    const float* xcb = xc + (size_t)sb * D_INNER * LL;
    for (int k = 0; k < D_INNER; k += 4) {
        const int kb = k + kq;
        v2f bb;
        bb.x = xcb[(size_t)kb       * LL + l0 + m];
        bb.y = xcb[(size_t)(kb + 1) * LL + l0 + m];
#pragma unroll
        for (int i = 0; i < 4; ++i) {
            v2f aa = *(const v2f*)(xpw + ((size_t)s * XDBL_E + i * 16 + m) * D_INNER + kb);
            acc[i] = wmma4(aa, bb, acc[i]);
        }
    }
    const int n  = lane & 15;
    const int mb = (lane >> 4) * 8;
    float* obase = xdbl + ((size_t)sb * LL + l0) * XDBL_E;
#pragma unroll
    for (int i = 0; i < 4; ++i)
#pragma unroll
        for (int rr = 0; rr < 8; ++rr)
            obase[(size_t)n * XDBL_E + i * 16 + mb + rr] = acc[i][rr];
}

// -------------------------------------------------------------------------
// Kernel 4: selective scan. 4 lanes per channel (4 states + 8 dt-ranks each),
// cross-lane sums via shfl_xor over the aligned 4-lane group. LDS chunks are
// double-buffered and filled by the Tensor Data Mover (TENSORcnt), with the
// TDM pad feature producing the conflict-free stride-17 u-tile. Gated y
// overwrites xc in place (branch-flipped coordinates preserved).
// -------------------------------------------------------------------------
__global__ __launch_bounds__(256) void k_scan(const float* __restrict__ zx,
                                              const float* __restrict__ xdbl,
                                              const float* __restrict__ dpw,
                                              const float* __restrict__ dbv,
                                              const float* __restrict__ Alog,
                                              const float* __restrict__ Dv,
                                              float* __restrict__ xc) {
    const int tid = threadIdx.x;
    const int wv  = tid >> 5;
    const int blk = blockIdx.x;          // 0..127
    const int sb  = blk >> 5;            // 32 blocks per (s,b)
    const int d0  = (blk & 31) * CPB;
    const int s   = sb >> 1;
    const int b   = sb & 1;

    const int ch  = tid >> 2;            // 0..63  channel within block
    const int sub = tid & 3;             // 0..3   lane within channel group
    const int d   = d0 + ch;

    __shared__ float sxd[2][CHUNK * XDBL_E];    // [buf][t*64 + e]
    __shared__ float sxc[2][CPB * (CHUNK + 1)]; // [buf][ch*17 + t]  (padded)

    float h[SPL], a2[SPL], wdt[RPL];
#pragma unroll
    for (int j = 0; j < SPL; ++j) {
        h[j]  = 0.0f;
        a2[j] = -__expf(Alog[((size_t)s * D_INNER + d) * D_STATE + sub * SPL + j])
                * 1.4426950408889634f;                 // pre-scale for exp2
    }
#pragma unroll
    for (int j = 0; j < RPL; ++j)
        wdt[j] = dpw[((size_t)s * D_INNER + d) * DT_RANK + sub * RPL + j];
    const float dbias = dbv[s * D_INNER + d];
    const float Dd    = Dv[s * D_INNER + d];

    const float* xdb   = xdbl + (size_t)sb * LL * XDBL_E;
    float* xcb         = xc + (size_t)sb * D_INNER * LL;
    const float* zbase = zx + ((size_t)b * (2 * D_INNER) + D_INNER) * LL;

    const int NCHUNK = LL / CHUNK;

#if USE_TDM
    if (wv == 0) {   // wave-level DMA: chunk 0 into buffer 0
        tdm_load((unsigned)(size_t)&sxd[0][0],
                 (unsigned long long)(size_t)xdb,
                 CHUNK * XDBL_E, 0, CHUNK * XDBL_E, false, 0);
        tdm_load((unsigned)(size_t)&sxc[0][0],
                 (unsigned long long)(size_t)(xcb + (size_t)d0 * LL),
                 CHUNK, CPB, LL, true, 3 /* pad 1 dw per 16 dw */);
    }
#endif

    for (int c = 0; c < NCHUNK; ++c) {
        const int cur = c & 1;
        const int t0  = c * CHUNK;
#if USE_TDM
        if (wv == 0) __builtin_amdgcn_s_wait_tensorcnt(0);
        __syncthreads();                       // data of chunk c visible to all
        if (wv == 0 && (c + 1) < NCHUNK) {     // prefetch chunk c+1 -> other buf
            const int tn = t0 + CHUNK;
            tdm_load((unsigned)(size_t)&sxd[1 - cur][0],
                     (unsigned long long)(size_t)(xdb + (size_t)tn * XDBL_E),
                     CHUNK * XDBL_E, 0, CHUNK * XDBL_E, false, 0);
            tdm_load((unsigned)(size_t)&sxc[1 - cur][0],
                     (unsigned long long)(size_t)(xcb + (size_t)d0 * LL + tn),
                     CHUNK, CPB, LL, true, 3);
        }
#else
        // fallback: cooperative synchronous staging
#pragma unroll
        for (int i = 0; i < (CHUNK * XDBL_E) / 256; ++i) {
            const int idx = i * 256 + tid;
            sxd[cur][idx] = xdb[(size_t)t0 * XDBL_E + idx];
        }
#pragma unroll
        for (int i = 0; i < (CPB * CHUNK) / 256; ++i) {
            const int idx = i * 256 + tid;
            const int di = idx >> 4, ti = idx & (CHUNK - 1);
            sxc[cur][di * (CHUNK + 1) + ti] = xcb[(size_t)(d0 + di) * LL + t0 + ti];
        }
        __syncthreads();
#endif

#pragma unroll 1
        for (int ti = 0; ti < CHUNK; ++ti) {
            const float* row = &sxd[cur][ti * XDBL_E];
            float dotp = 0.0f;
#pragma unroll
            for (int j = 0; j < RPL; ++j) dotp += row[sub * RPL + j] * wdt[j];
            dotp += __shfl_xor(dotp, 1, 32);
            dotp += __shfl_xor(dotp, 2, 32);
            const float dot   = dotp + dbias;
            const float delta = (dot > 20.0f) ? dot : log1pf(__expf(dot));
            const float u  = sxc[cur][ch * (CHUNK + 1) + ti];
            const float du = delta * u;
            float yp = 0.0f;
#pragma unroll
            for (int j = 0; j < SPL; ++j) {
                const float dA = fast_exp2(delta * a2[j]);
                h[j] = dA * h[j] + du * row[DT_RANK + sub * SPL + j];
                yp += h[j] * row[DT_RANK + D_STATE + sub * SPL + j];
            }
            yp += __shfl_xor(yp, 1, 32);
            yp += __shfl_xor(yp, 2, 32);
            if (sub == 0)
                sxc[cur][ch * (CHUNK + 1) + ti] = yp + Dd * u;
        }
        __syncthreads();

        // gated writeout (z = second half of zx, branch-flipped index)
#pragma unroll
        for (int i = 0; i < (CPB * CHUNK) / 256; ++i) {
            const int idx = i * 256 + tid;
            const int di = idx >> 4, ti = idx & (CHUNK - 1);
            const int lp = t0 + ti;
            const int lg = s ? (LL - 1 - lp) : lp;
            const float zv = zbase[(size_t)(d0 + di) * LL + lg];
            const float g  = zv / (1.0f + __expf(-zv));
            xcb[(size_t)(d0 + di) * LL + lp] = sxc[cur][di * (CHUNK + 1) + ti] * g;
        }
        __syncthreads();
    }
}

// -------------------------------------------------------------------------
// Kernel 5: out[b,l,o] = sum_d 0.5*(yf[b,d,l] + yb[b,d,L-1-l]) * wo[o,d]
// -------------------------------------------------------------------------
__global__ __launch_bounds__(256) void k_out(const float* __restrict__ y,
                                             const float* __restrict__ wo,
                                             float* __restrict__ out) {
    const int lane = threadIdx.x & 31;
    const int wv   = threadIdx.x >> 5;
    const int W    = blockIdx.x * 8 + wv;
    const int OT   = D_MODEL / 32;               // 16
    const int LT   = LL / 32;                    // 128
    const int b    = W / (OT * LT);
    const int r    = W % (OT * LT);
    const int o0   = (r / LT) * 32;
    const int l0   = (r % LT) * 32;

    const int m  = lane & 15;
    const int kq = (lane >> 4) * 2;

    const float* yf = y + (size_t)(0 * NB + b) * D_INNER * LL;
    const float* yb = y + (size_t)(1 * NB + b) * D_INNER * LL;
    const int lA = l0 + m;
    const int lB = l0 + 16 + m;

    v8f a00 = {}, a01 = {}, a10 = {}, a11 = {};
    for (int k = 0; k < D_INNER; k += 4) {
        const int kb = k + kq;
        v2f wa0 = *(const v2f*)(wo + (size_t)(o0 + m)      * D_INNER + kb);
        v2f wa1 = *(const v2f*)(wo + (size_t)(o0 + 16 + m) * D_INNER + kb);
        v2f b0, b1;
        b0.x = 0.5f * (yf[(size_t)kb       * LL + lA] + yb[(size_t)kb       * LL + (LL - 1 - lA)]);
        b0.y = 0.5f * (yf[(size_t)(kb + 1) * LL + lA] + yb[(size_t)(kb + 1) * LL + (LL - 1 - lA)]);
        b1.x = 0.5f * (yf[(size_t)kb       * LL + lB] + yb[(size_t)kb       * LL + (LL - 1 - lB)]);
        b1.y = 0.5f * (yf[(size_t)(kb + 1) * LL + lB] + yb[(size_t)(kb + 1) * LL + (LL - 1 - lB)]);
        a00 = wmma4(wa0, b0, a00);
        a01 = wmma4(wa0, b1, a01);
        a10 = wmma4(wa1, b0, a10);
        a11 = wmma4(wa1, b1, a11);
    }
    const int n  = lane & 15;
    const int mb = (lane >> 4) * 8;
#pragma unroll
    for (int rr = 0; rr < 8; ++rr) {
        const int mm = mb + rr;
        out[((size_t)b * LL + l0 + n)      * D_MODEL + o0 + mm]      = a00[rr];
        out[((size_t)b * LL + l0 + 16 + n) * D_MODEL + o0 + mm]      = a01[rr];
        out[((size_t)b * LL + l0 + n)      * D_MODEL + o0 + 16 + mm] = a10[rr];
        out[((size_t)b * LL + l0 + 16 + n) * D_MODEL + o0 + 16 + mm] = a11[rr];
    }
}

// -------------------------------------------------------------------------
extern "C" void kernel_launch(void* const* d_in, const int* in_sizes, int n_in,
                              void* d_out, int out_size, void* d_ws, size_t ws_size,
                              hipStream_t stream) {
    const float* hs    = (const float*)d_in[0];   // (B, L, 512)
    const float* ipw   = (const float*)d_in[1];   // (4096, 512)
    const float* cw    = (const float*)d_in[2];   // (2, 2048, 4)
    const float* cb    = (const float*)d_in[3];   // (2, 2048)
    const float* xpw   = (const float*)d_in[4];   // (2, 64, 2048)
    const float* dpw   = (const float*)d_in[5];   // (2, 2048, 32)
    const float* dbias = (const float*)d_in[6];   // (2, 2048)
    const float* Alog  = (const float*)d_in[7];   // (2, 2048, 16)
    const float* Dv    = (const float*)d_in[8];   // (2, 2048)
    const float* wo    = (const float*)d_in[9];   // (512, 2048)
    float* out = (float*)d_out;
    float* ws  = (float*)d_ws;

    float* zx   = ws;                                     // 2*4096*4096 f
    float* xc   = zx + (size_t)NB * 2 * D_INNER * LL;     // 2*2*2048*4096 f
    float* xdbl = xc + (size_t)2 * NB * D_INNER * LL;     // 2*2*4096*64 f

    k_inproj<<<4096, 256, 0, stream>>>(hs, ipw, zx);
    k_conv<<<(unsigned)(((size_t)2 * NB * D_INNER * LL) / 256), 256, 0, stream>>>(zx, cw, cb, xc);
    k_xdbl<<<128, 256, 0, stream>>>(xc, xpw, xdbl);
    k_scan<<<128, 256, 0, stream>>>(zx, xdbl, dpw, dbias, Alog, Dv, xc);
    k_out<<<512, 256, 0, stream>>>(xc, wo, out);
}